// Overlaps_592705487146
// MI455X (gfx1250) — compile-verified
//
#include <hip/hip_runtime.h>
#include <hip/hip_bf16.h>

typedef unsigned int       u32;
typedef unsigned long long u64;
typedef float v4f __attribute__((ext_vector_type(4)));
typedef u32   v4u __attribute__((ext_vector_type(4)));
typedef u32   v8u __attribute__((ext_vector_type(8)));

#define TILE_I      128        // rows of boxes0 per block
#define BLOCK_T     256        // threads per block
#define COLS_PER_T  4          // columns (boxes1) per thread -> 16B NT store
#define TILE_J      (BLOCK_T * COLS_PER_T)   // 1024 columns per block

__global__ __launch_bounds__(BLOCK_T)
void Overlaps_592705487146_kernel(const float* __restrict__ boxes0,
                                  const int*   __restrict__ labels0,
                                  const int*   __restrict__ batches0,
                                  const float* __restrict__ boxes1,
                                  const int*   __restrict__ labels1,
                                  const int*   __restrict__ batches1,
                                  float* __restrict__ out,
                                  int n0, int n1) {
    __shared__ float sBox[TILE_I * 4];   // boxes0 tile, DMA'd via TDM
    __shared__ float sArea[TILE_I];
    __shared__ int   sKey[TILE_I];

    const int rowBase = blockIdx.y * TILE_I;
    const int nrows   = min(TILE_I, n0 - rowBase);
    const int t       = threadIdx.x;

    // ---- Stage boxes0 tile (nrows x 4 f32, contiguous) into LDS via the
    // ---- Tensor Data Mover. Wave 0 issues one 1-D TDM descriptor.
    if (t < 32) {
        const u64 ga    = (u64)(const void*)(boxes0 + (size_t)rowBase * 4);
        const u32 lds   = (u32)(u64)(const void*)&sBox[0];   // low 32 bits = LDS byte offset
        const u32 nelem = (u32)nrows * 4;                    // f32 elements

        v4u g0;
        g0[0] = 1u;                                          // count=1 (valid), no gather
        g0[1] = lds;                                         // lds_addr
        g0[2] = (u32)ga;                                     // global_addr[31:0]
        g0[3] = (u32)((ga >> 32) & 0x01FFFFFFu) | (2u << 30);// global_addr[56:32] | type=2

        v8u g1;
        g1[0] = (2u << 16);                 // workgroup_mask=0, data_size=2 (4 bytes)
        g1[1] = (nelem & 0xFFFFu) << 16;    // tensor_dim0[15:0] (bits 63:48)
        g1[2] = ((nelem >> 16) & 0xFFFFu)   // tensor_dim0[31:16]
              | (1u << 16);                 // tensor_dim1[15:0] = 1
        g1[3] = (nelem << 16);              // tensor_dim1[31:16]=0 | tile_dim0 = nelem
        g1[4] = 1u;                         // tile_dim1 = 1, tile_dim2 = 0
        g1[5] = nelem;                      // tensor_dim0_stride[31:0]
        g1[6] = 0u;                         // stride0[47:32]=0, tensor_dim1_stride[15:0]=0
        g1[7] = 0u;                         // tensor_dim1_stride[47:16]=0

        asm volatile("tensor_load_to_lds %0, %1" :: "s"(g0), "s"(g1) : "memory");
        __builtin_amdgcn_s_wait_tensorcnt(0);
    }
    __syncthreads();

    // ---- Per-row key / area prep (tiny, cooperative)
    if (t < TILE_I) {
        if (t < nrows) {
            const float bx1 = sBox[t * 4 + 0], by1 = sBox[t * 4 + 1];
            const float bx2 = sBox[t * 4 + 2], by2 = sBox[t * 4 + 3];
            sArea[t] = (bx2 - bx1) * (by2 - by1);
            sKey[t]  = (labels0[rowBase + t] << 8) | batches0[rowBase + t];
        } else {
            sArea[t] = 0.0f;
            sKey[t]  = -1;
        }
    }
    __syncthreads();

    // ---- Per-thread column state (4 consecutive columns)
    const int c0 = blockIdx.x * TILE_J + t * COLS_PER_T;
    if (c0 >= n1) return;

    v4f   cb[COLS_PER_T];
    float ca[COLS_PER_T];
    int   ck[COLS_PER_T];
#pragma unroll
    for (int k = 0; k < COLS_PER_T; ++k) {
        const int c = c0 + k;
        cb[k] = *(const v4f*)(boxes1 + 4 * (size_t)c);       // x1,y1,x2,y2
        ca[k] = (cb[k].z - cb[k].x) * (cb[k].w - cb[k].y);
        ck[k] = (labels1[c] << 8) | batches1[c];
    }

    const bool full4 = (c0 + COLS_PER_T) <= n1;
    float* orow = out + (size_t)rowBase * (size_t)n1 + (size_t)c0;

    for (int r = 0; r < nrows; ++r) {
        const int rk = sKey[r];
        v4f res = (v4f){0.f, 0.f, 0.f, 0.f};

        const bool m = (rk == ck[0]) | (rk == ck[1]) | (rk == ck[2]) | (rk == ck[3]);
        // wave-uniform early-out
        if (__builtin_amdgcn_ballot_w32(m) != 0u) {
            const float rx1 = sBox[r * 4 + 0], ry1 = sBox[r * 4 + 1];
            const float rx2 = sBox[r * 4 + 2], ry2 = sBox[r * 4 + 3];
            const float ra  = sArea[r];
#pragma unroll
            for (int k = 0; k < COLS_PER_T; ++k) {
                const float ix1 = fmaxf(rx1, cb[k].x);
                const float iy1 = fmaxf(ry1, cb[k].y);
                const float ix2 = fminf(rx2, cb[k].z);
                const float iy2 = fminf(ry2, cb[k].w);
                const float iw  = fmaxf(ix2 - ix1, 0.0f);
                const float ih  = fmaxf(iy2 - iy1, 0.0f);
                const float inter = iw * ih;
                const float uni   = ra + ca[k] - inter;
                // exact IEEE fp32 division (matches jnp inter/union); only on
                // the rare match path so it costs nothing on the store roofline
                float iou = inter / uni;
                iou = (uni > 0.0f) ? iou : 0.0f;
                res[k] = (rk == ck[k]) ? iou : 0.0f;
            }
        }

        if (full4) {
            __builtin_nontemporal_store(res, (v4f*)orow);   // 16B streaming store
        } else {
#pragma unroll
            for (int k = 0; k < COLS_PER_T; ++k)
                if (c0 + k < n1) __builtin_nontemporal_store(res[k], orow + k);
        }
        orow += n1;
    }
}

extern "C" void kernel_launch(void* const* d_in, const int* in_sizes, int n_in,
                              void* d_out, int out_size, void* d_ws, size_t ws_size,
                              hipStream_t stream) {
    const float* boxes0   = (const float*)d_in[0];
    const int*   labels0  = (const int*)  d_in[1];
    const int*   batches0 = (const int*)  d_in[2];
    const float* boxes1   = (const float*)d_in[3];
    const int*   labels1  = (const int*)  d_in[4];
    const int*   batches1 = (const int*)  d_in[5];
    float* out = (float*)d_out;

    const int n0 = in_sizes[1];   // N0 (labels0 count)
    const int n1 = in_sizes[4];   // N1 (labels1 count)

    dim3 grid((n1 + TILE_J - 1) / TILE_J, (n0 + TILE_I - 1) / TILE_I);
    dim3 block(BLOCK_T);
    Overlaps_592705487146_kernel<<<grid, block, 0, stream>>>(
        boxes0, labels0, batches0, boxes1, labels1, batches1, out, n0, n1);
}